// LU_Unit_2327872274943
// MI455X (gfx1250) — compile-verified
//
#include <hip/hip_runtime.h>

// Upsample2x(nearest) + ReplicationPad(1) + depthwise 3x3 conv + bias.
// B=8, C=128, H=W=128 -> out [8,128,256,256], all f32.
//
// Bandwidth-bound (67 MB in + 268 MB out ~= 14.4 us at 23.3 TB/s).
// Each output pixel depends on a 2x2 input neighborhood; the 3x3 weights
// collapse into 16 per-channel phase coefficients. Input tiles are staged
// into LDS with the CDNA5 async copy engine (global_load_async_to_lds_b128,
// tracked by ASYNCcnt) so halo reuse is served from LDS, and outputs are
// written with non-temporal hints (256 MB stream > 192 MB L2).

#define IN_H 128
#define IN_W 128
#define IN_C 128
#define TILE_ROWS 8
#define HALO_ROWS (TILE_ROWS + 2)   // 10 rows in LDS
#define OUT_W (2 * IN_W)

typedef float v2f __attribute__((ext_vector_type(2)));

__global__ __launch_bounds__(256)
void upblur_kernel(const float* __restrict__ x,
                   const float* __restrict__ w,
                   const float* __restrict__ bias,
                   float* __restrict__ out)
{
    __shared__ float tile[HALO_ROWS * IN_W];   // 5120 bytes

    const int chunk = blockIdx.x;              // 0..15 : row chunk
    const int n     = blockIdx.y;              // 0..1023 : b*C + c
    const int c     = n & (IN_C - 1);
    const int tid   = threadIdx.x;             // 0..255 (8 waves)
    const int r0    = chunk * TILE_ROWS;

    const float* __restrict__ src = x + (size_t)n * (IN_H * IN_W);

    // ------------------------------------------------------------------
    // Async copy: rows [r0-1, r0+8] (edge-clamped) -> LDS, 16B per lane.
    // 10 rows * 32 float4 = 320 chunks over 256 threads (<=2 each).
    // ------------------------------------------------------------------
    const unsigned lds_base = (unsigned)(unsigned long long)(&tile[0]);
    for (int idx = tid; idx < HALO_ROWS * (IN_W / 4); idx += 256) {
        int t   = idx >> 5;                    // tile row 0..9
        int vec = idx & 31;                    // float4 within row
        int gr  = r0 - 1 + t;                  // global input row (clamped)
        gr = gr < 0 ? 0 : (gr > IN_H - 1 ? IN_H - 1 : gr);
        unsigned goff = (unsigned)(gr * (IN_W * 4) + vec * 16);  // byte off
        unsigned loff = lds_base + (unsigned)(t * (IN_W * 4) + vec * 16);
        asm volatile("global_load_async_to_lds_b128 %0, %1, %2"
                     :: "v"(loff), "v"(goff), "s"(src)
                     : "memory");
    }
#if __has_builtin(__builtin_amdgcn_s_wait_asynccnt)
    __builtin_amdgcn_s_wait_asynccnt(0);
#else
    asm volatile("s_wait_asynccnt 0x0" ::: "memory");
#endif
    __syncthreads();

    // ------------------------------------------------------------------
    // Fold 2x nearest upsample + edge pad into 16 per-channel coefficients
    // (4 per output phase). Derived from the actual 3x3 weights, so this
    // holds for any depthwise kernel, not just the binomial blur.
    //   even index: contributions from {i-1 (kernel row 0), i (rows 1+2)}
    //   odd  index: contributions from {i (rows 0+1), i+1 (row 2)}
    // ------------------------------------------------------------------
    const float* wc = w + c * 9;               // OIHW, I=1 -> 9 per channel
    const float w00 = wc[0], w01 = wc[1], w02 = wc[2];
    const float w10 = wc[3], w11 = wc[4], w12 = wc[5];
    const float w20 = wc[6], w21 = wc[7], w22 = wc[8];

    // phase (even row, even col): x[i-1][j-1], x[i-1][j], x[i][j-1], x[i][j]
    const float A0 = w00,          A1 = w01 + w02;
    const float A2 = w10 + w20,    A3 = w11 + w12 + w21 + w22;
    // phase (even row, odd col):  x[i-1][j], x[i-1][j+1], x[i][j], x[i][j+1]
    const float B0 = w00 + w01,    B1 = w02;
    const float B2 = w10 + w11 + w20 + w21, B3 = w12 + w22;
    // phase (odd row, even col):  x[i][j-1], x[i][j], x[i+1][j-1], x[i+1][j]
    const float C0 = w00 + w10,    C1 = w01 + w02 + w11 + w12;
    const float C2 = w20,          C3 = w21 + w22;
    // phase (odd row, odd col):   x[i][j], x[i][j+1], x[i+1][j], x[i+1][j+1]
    const float D0 = w00 + w01 + w10 + w11, D1 = w02 + w12;
    const float D2 = w20 + w21,             D3 = w22;

    const float bb = bias[c];

    // ------------------------------------------------------------------
    // Compute: 2 rows per pass x 4 passes covers 8 input rows.
    // Lane j produces the 2x2 output block at (2i, 2j) -> float2 stores
    // are lane-contiguous (wave writes 256B spans per output row).
    // ------------------------------------------------------------------
    const int j    = tid & (IN_W - 1);         // input column 0..127
    const int rofs = tid >> 7;                 // 0 or 1
    const int jm   = (j == 0) ? 0 : j - 1;
    const int jp   = (j == IN_W - 1) ? IN_W - 1 : j + 1;

    float* __restrict__ dst = out + (size_t)n * (2 * IN_H * OUT_W);

    for (int it = 0; it < 4; ++it) {
        const int rl = rofs + 2 * it;          // local input row 0..7
        const float* t0 = &tile[(rl    ) * IN_W];  // row i-1 (clamped)
        const float* t1 = &tile[(rl + 1) * IN_W];  // row i
        const float* t2 = &tile[(rl + 2) * IN_W];  // row i+1 (clamped)

        const float x00 = t0[jm], x01 = t0[j], x02 = t0[jp];
        const float x10 = t1[jm], x11 = t1[j], x12 = t1[jp];
        const float x20 = t2[jm], x21 = t2[j], x22 = t2[jp];

        const float o00 = A0 * x00 + A1 * x01 + A2 * x10 + A3 * x11 + bb;
        const float o01 = B0 * x01 + B1 * x02 + B2 * x11 + B3 * x12 + bb;
        const float o10 = C0 * x10 + C1 * x11 + C2 * x20 + C3 * x21 + bb;
        const float o11 = D0 * x11 + D1 * x12 + D2 * x21 + D3 * x22 + bb;

        const int orow = 2 * (r0 + rl);
        v2f e = {o00, o01};
        v2f o = {o10, o11};
        __builtin_nontemporal_store(e, (v2f*)(dst + (size_t)orow * OUT_W + 2 * j));
        __builtin_nontemporal_store(o, (v2f*)(dst + (size_t)(orow + 1) * OUT_W + 2 * j));
    }
}

extern "C" void kernel_launch(void* const* d_in, const int* in_sizes, int n_in,
                              void* d_out, int out_size, void* d_ws, size_t ws_size,
                              hipStream_t stream) {
    const float* x = (const float*)d_in[0];   // [8,128,128,128] f32
    const float* w = (const float*)d_in[1];   // [128,1,3,3] f32
    const float* b = (const float*)d_in[2];   // [128] f32
    float* out     = (float*)d_out;           // [8,128,256,256] f32
    (void)in_sizes; (void)n_in; (void)out_size; (void)d_ws; (void)ws_size;

    dim3 grid(IN_H / TILE_ROWS, 8 * IN_C);    // (16, 1024)
    upblur_kernel<<<grid, 256, 0, stream>>>(x, w, b, out);
}